// TDrumorGCN_29111288332558
// MI455X (gfx1250) — compile-verified
//
#include <hip/hip_runtime.h>

// GCN forward on gfx1250: fp32 WMMA (V_WMMA_F32_16X16X4_F32) for the dense
// GEMMs (exact fp32, matches reference numerics), atomic scatter-add for the
// message passing (L2-resident: 25.6MB feature matrix << 192MB L2).
//
// LDS layout tuned for the WMMA fragment access pattern:
//  - X tile row stride padded to 68 floats -> A-frag b64 loads conflict-free
//  - W chunk stored as interleaved (k,k+1) pairs, pair-row stride 288 floats
//    -> B-frag is one aligned ds_load_b64, lanes 0-15 / 16-31 hit disjoint
//    bank halves (288*4 mod 256B == 128B == 32 banks)

typedef __attribute__((ext_vector_type(2))) float v2f;
typedef __attribute__((ext_vector_type(8))) float v8f;

#define NCOL 128   // HID == OUT_FEATS == 128 in the reference
#define TILE_M 128 // rows of C per block (8 waves x 16)
#define KC 64      // K-chunk staged in LDS (K % KC == 0 for K=512,128)
#define LXS 68     // lx row stride (floats): 272B, 16B aligned, bank-spread
#define LWS 288    // lw2 k-pair row stride (floats): odd rows +32 banks

// ---------------------------------------------------------------------------
// C[M x 128] = X[M x K] * W[K x 128], fp32, WMMA 16x16x4
// ---------------------------------------------------------------------------
__global__ __launch_bounds__(256) void k_gemm_wmma(
    const float* __restrict__ X, const float* __restrict__ W,
    float* __restrict__ C, int M, int K) {
  __shared__ float lw2[(KC / 2) * LWS];  // 36KB: W chunk, (k-pair, col*2+{0,1})
  __shared__ float lx[TILE_M * LXS];     // 34KB: X chunk, [128][KC] padded

  const int tid = threadIdx.x;
  const int lane = tid & 31;
  const int wave = tid >> 5;  // 0..7 -> 16-row strip per wave
  const int mblock = blockIdx.x * TILE_M;
  const int mstrip = mblock + wave * 16;

  v8f acc[8] = {};  // 8 tiles of 16x16 covering all 128 output cols

  const int lane15 = lane & 15;
  const int ksel = (lane < 16) ? 0 : 2;  // A/B K sub-select per lane half
  const int mrow = wave * 16 + lane15;   // row within the LDS X tile

  // Per-lane base pointers into LDS for the compute loop
  const float* axp = lx + mrow * LXS + ksel;
  const float* bbp = lw2 + (ksel >> 1) * LWS + lane15 * 2;

  for (int k0 = 0; k0 < K; k0 += KC) {
    __syncthreads();
    // Stage W chunk as interleaved (k,k+1) pairs: lw2[kk2*LWS + c*2 + (k&1)]
    for (int i = tid; i < (KC / 2) * (NCOL / 4); i += 256) {  // 1024 items
      const int kk2 = i >> 5;       // k-pair row 0..31
      const int c4 = i & 31;        // col quad 0..31
      const float* wr = W + (size_t)(k0 + kk2 * 2) * NCOL + c4 * 4;
      const float4 w0 = *(const float4*)wr;
      const float4 w1 = *(const float4*)(wr + NCOL);
      float* dst = lw2 + kk2 * LWS + c4 * 8;
      ((float2*)dst)[0] = make_float2(w0.x, w1.x);
      ((float2*)dst)[1] = make_float2(w0.y, w1.y);
      ((float2*)dst)[2] = make_float2(w0.z, w1.z);
      ((float2*)dst)[3] = make_float2(w0.w, w1.w);
    }
    // Stage X chunk: TILE_M rows x KC cols (row stride K in global)
    for (int i = tid; i < TILE_M * (KC / 4); i += 256) {  // 2048 items
      const int r = i >> 4;
      const int c4 = i & 15;
      const int gr = mblock + r;
      float4 v = make_float4(0.f, 0.f, 0.f, 0.f);
      if (gr < M) v = *(const float4*)(X + (size_t)gr * K + k0 + c4 * 4);
      *(float4*)(lx + r * LXS + c4 * 4) = v;
    }
    __syncthreads();

    // 16 k-steps of 4, 8 column tiles each -> 128 wmma per chunk per wave
#pragma unroll 4
    for (int kk = 0; kk < KC; kk += 4) {
      const v2f a = *(const v2f*)(axp + kk);
      const float* brow = bbp + (kk >> 1) * LWS;
#pragma unroll
      for (int t = 0; t < 8; ++t) {
        const v2f b = *(const v2f*)(brow + t * 32);
        acc[t] = __builtin_amdgcn_wmma_f32_16x16x4_f32(
            /*neg_a=*/false, a, /*neg_b=*/false, b,
            /*c_mod=*/(short)0, acc[t], /*reuse_a=*/false, /*reuse_b=*/false);
      }
    }
  }

  // Store: VGPR v -> row (v | v+8) per lane half, col = tile*16 + lane15
  if (mstrip < M) {
    const int mo = (lane < 16) ? 0 : 8;
#pragma unroll
    for (int t = 0; t < 8; ++t) {
#pragma unroll
      for (int v = 0; v < 8; ++v) {
        const int m = mstrip + mo + v;
        if (m < M) C[(size_t)m * NCOL + t * 16 + lane15] = acc[t][v];
      }
    }
  }
}

// ---------------------------------------------------------------------------
// Graph-side kernels
// ---------------------------------------------------------------------------
__global__ void k_deg_init(float* deg, int N) {
  int i = blockIdx.x * blockDim.x + threadIdx.x;
  if (i < N) deg[i] = 1.0f;  // self-loop contribution
}

__global__ void k_deg_count(const int* __restrict__ col, float* deg, int E) {
  int i = blockIdx.x * blockDim.x + threadIdx.x;
  if (i < E) atomicAdd(&deg[col[i]], 1.0f);
}

__global__ void k_rsqrt(float* deg, int N) {
  int i = blockIdx.x * blockDim.x + threadIdx.x;
  if (i < N) {
    float d = deg[i];
    deg[i] = (d > 0.f) ? rsqrtf(d) : 0.f;
  }
}

// agg[n,f] = hlin[n,f] * dinv[n]^2   (self-loop term; also zero-initializes)
__global__ void k_selfloop_init(const float* __restrict__ hlin,
                                const float* __restrict__ dinv,
                                float* __restrict__ agg, int N) {
  int i = blockIdx.x * blockDim.x + threadIdx.x;
  if (i < N * NCOL) {
    int n = i >> 7;
    float w = dinv[n];
    agg[i] = hlin[i] * w * w;
  }
}

// agg[dst,f] += hlin[src,f] * dinv[src]*dinv[dst]; 2 edges per 256-thread block
__global__ __launch_bounds__(256) void k_edge_scatter(
    const float* __restrict__ hlin, const int* __restrict__ row,
    const int* __restrict__ col, const float* __restrict__ dinv,
    float* __restrict__ agg, int E) {
  int e = blockIdx.x * 2 + (threadIdx.x >> 7);
  int f = threadIdx.x & 127;
  if (e >= E) return;
  int s = row[e];
  int d = col[e];
  float w = dinv[s] * dinv[d];
  atomicAdd(&agg[(size_t)d * NCOL + f], hlin[(size_t)s * NCOL + f] * w);
}

__global__ void k_bias_act(float* __restrict__ h, const float* __restrict__ b,
                           int N, int do_relu) {
  int i = blockIdx.x * blockDim.x + threadIdx.x;
  if (i < N * NCOL) {
    float v = h[i] + b[i & 127];
    h[i] = (do_relu && v < 0.f) ? 0.f : v;
  }
}

__global__ void k_zero(float* p, int n) {
  int i = blockIdx.x * blockDim.x + threadIdx.x;
  if (i < n) p[i] = 0.f;
}

// hs[batch[n], f] += h[n, f]
__global__ void k_pool(const float* __restrict__ h, const int* __restrict__ batch,
                       float* __restrict__ hs, int N) {
  int i = blockIdx.x * blockDim.x + threadIdx.x;
  if (i < N * NCOL) {
    int n = i >> 7;
    atomicAdd(&hs[(size_t)batch[n] * NCOL + (i & 127)], h[i]);
  }
}

// ---------------------------------------------------------------------------
extern "C" void kernel_launch(void* const* d_in, const int* in_sizes, int n_in,
                              void* d_out, int out_size, void* d_ws, size_t ws_size,
                              hipStream_t stream) {
  const float* x = (const float*)d_in[0];
  const float* W1 = (const float*)d_in[1];
  const float* b1 = (const float*)d_in[2];
  const float* W2 = (const float*)d_in[3];
  const float* b2 = (const float*)d_in[4];
  const int* eidx = (const int*)d_in[5];
  const int* batch = (const int*)d_in[6];

  const int N = in_sizes[6];          // 50000 nodes
  const int E = in_sizes[5] / 2;      // 640000 edges
  const int K1 = in_sizes[0] / N;     // 512 input feats
  const int* row = eidx;              // sources
  const int* col = eidx + E;          // targets

  // Workspace: dinv [N] | bufA [N*128] | bufB [N*128]  (~51.4 MB)
  float* ws = (float*)d_ws;
  float* dinv = ws;
  float* bufA = ws + (((size_t)N + 63) & ~(size_t)63);
  float* bufB = bufA + (size_t)N * NCOL;

  // Output: [hs (G*128)] [h (N*128)]
  const int G = out_size / NCOL - N;  // 256 graphs
  float* hs = (float*)d_out;
  float* hout = hs + (size_t)G * NCOL;

  const dim3 b256(256);
  const int nfBlocks = (N * NCOL + 255) / 256;
  const int gemmBlocks = (N + TILE_M - 1) / TILE_M;
  const int edgeBlocks = (E + 1) / 2;

  // Degree / normalization
  k_deg_init<<<(N + 255) / 256, b256, 0, stream>>>(dinv, N);
  k_deg_count<<<(E + 255) / 256, b256, 0, stream>>>(col, dinv, E);
  k_rsqrt<<<(N + 255) / 256, b256, 0, stream>>>(dinv, N);

  // Layer 1: h1 = relu(Ahat * (x@W1) + b1)
  k_gemm_wmma<<<gemmBlocks, b256, 0, stream>>>(x, W1, bufA, N, K1);
  k_selfloop_init<<<nfBlocks, b256, 0, stream>>>(bufA, dinv, bufB, N);
  k_edge_scatter<<<edgeBlocks, b256, 0, stream>>>(bufA, row, col, dinv, bufB, E);
  k_bias_act<<<nfBlocks, b256, 0, stream>>>(bufB, b1, N, 1);

  // Layer 2: h2 = Ahat * (h1@W2) + b2  (written straight into d_out's h region)
  k_gemm_wmma<<<gemmBlocks, b256, 0, stream>>>(bufB, W2, bufA, N, NCOL);
  k_selfloop_init<<<nfBlocks, b256, 0, stream>>>(bufA, dinv, hout, N);
  k_edge_scatter<<<edgeBlocks, b256, 0, stream>>>(bufA, row, col, dinv, hout, E);
  k_bias_act<<<nfBlocks, b256, 0, stream>>>(hout, b2, N, 0);

  // Global add pool
  k_zero<<<(G * NCOL + 255) / 256, b256, 0, stream>>>(hs, G * NCOL);
  k_pool<<<nfBlocks, b256, 0, stream>>>(hout, batch, hs, N);
}